// yoloLoss_78030965834119
// MI455X (gfx1250) — compile-verified
//
#include <hip/hip_runtime.h>
#include <cstdint>

// ---------------------------------------------------------------------------
// YOLO loss on MI455X (gfx1250).
// Memory-bound: ~193 MB in, 4 B out -> HBM roofline ~8.3 us @ 23.3 TB/s.
// No matmul structure => no WMMA; use CDNA5 async global->LDS streaming path
// (global_load_async_to_lds_b128 + s_wait_asynccnt, ASYNCcnt-tracked).
// ---------------------------------------------------------------------------

#define SGRID 14
#define CELL_FLOATS 30
#define BLOCK 256
#define F4_PER_TENSOR (BLOCK * CELL_FLOATS / 4)   // 1920 float4 per tensor per block
#define F4_TOTAL (2 * F4_PER_TENSOR)              // 3840
#define LOADS_PER_THREAD (F4_TOTAL / BLOCK)       // 15

#ifndef USE_ASYNC
#define USE_ASYNC 1
#endif

#if USE_ASYNC
// CDNA5 async bulk copy: per-lane 16B global -> LDS, tracked by ASYNCcnt.
__device__ __forceinline__ void async_ld_b128(uint32_t lds_off, uint64_t gaddr) {
  asm volatile("global_load_async_to_lds_b128 %0, %1, off"
               :: "v"(lds_off), "v"(gaddr)
               : "memory");
}
__device__ __forceinline__ void wait_async0() {
  asm volatile("s_wait_asynccnt 0" ::: "memory");
}
#endif

// 1/14 rounded to f32; <=1 ulp vs IEEE division, tolerance-safe.
#define INV14 (1.0f / 14.0f)

__global__ __launch_bounds__(BLOCK)
void yolo_loss_stage1(const float* __restrict__ pred,
                      const float* __restrict__ targ,
                      float* __restrict__ partials,
                      int ncells) {
  __shared__ float4 tiles[F4_TOTAL];          // [0,1920): pred, [1920,3840): target
  __shared__ float red[BLOCK / 32];

  const int tid = threadIdx.x;
  const int blk = blockIdx.x;
  // last fully-valid float4 index within each tensor (32-bit: max ~6M)
  const int maxf4 = (ncells * CELL_FLOATS) / 4 - 1;

#if USE_ASYNC
  {
    const uint32_t lds_base = (uint32_t)(uintptr_t)(void*)tiles; // low 32b = LDS offset
    #pragma unroll
    for (int i = 0; i < LOADS_PER_THREAD; ++i) {
      const int gidx = i * BLOCK + tid;                 // 0..3839
      const bool is_pred = (gidx < F4_PER_TENSOR);
      int f4i = blk * F4_PER_TENSOR + (is_pred ? gidx : gidx - F4_PER_TENSOR);
      if (f4i > maxf4) f4i = maxf4;                     // clamp (never hit for 4096x14x14)
      const float* base = is_pred ? pred : targ;
      async_ld_b128(lds_base + (uint32_t)gidx * 16u,
                    (uint64_t)(uintptr_t)(base + (size_t)f4i * 4));
    }
    wait_async0();
  }
#else
  {
    const float4* gp = (const float4*)pred;
    const float4* gt = (const float4*)targ;
    #pragma unroll
    for (int i = 0; i < LOADS_PER_THREAD; ++i) {
      const int gidx = i * BLOCK + tid;
      const bool is_pred = (gidx < F4_PER_TENSOR);
      int f4i = blk * F4_PER_TENSOR + (is_pred ? gidx : gidx - F4_PER_TENSOR);
      if (f4i > maxf4) f4i = maxf4;
      tiles[gidx] = is_pred ? gp[f4i] : gt[f4i];
    }
  }
#endif
  __syncthreads();

  // ---- per-cell loss (stride-30 LDS reads: 32 lanes -> 32 distinct banks) ----
  const int cell = blk * BLOCK + tid;
  const float* sp = (const float*)tiles;
  const float* p = sp + tid * CELL_FLOATS;                        // pred cell (30 f)
  const float* t = sp + F4_PER_TENSOR * 4 + tid * CELL_FLOATS;    // target cell (30 f)

  float val = 0.0f;
  if (cell < ncells) {
    const float conf = t[4];
    if (conf == 0.0f) {                       // no-object term, weight 0.5
      const float d4 = p[4] - t[4];
      const float d9 = p[9] - t[9];
      val = 0.5f * (d4 * d4 + d9 * d9);
    }
    if (conf > 0.0f) {                        // object cell
      // target box 0 -> xyxy (xy/14, +-wh/2), mirror reference op order
      const float tx = t[0] * INV14, ty = t[1] * INV14;
      const float tw = t[2], th = t[3];
      const float tx1 = tx - 0.5f * tw, ty1 = ty - 0.5f * th;
      const float tx2 = tx + 0.5f * tw, ty2 = ty + 0.5f * th;
      const float ta = (tx2 - tx1) * (ty2 - ty1);

      float iou0 = 0.0f, iou1 = 0.0f;
      #pragma unroll
      for (int b = 0; b < 2; ++b) {
        const float* pb = p + 5 * b;
        const float px = pb[0] * INV14, py = pb[1] * INV14;
        const float pw = pb[2], ph = pb[3];
        const float px1 = px - 0.5f * pw, py1 = py - 0.5f * ph;
        const float px2 = px + 0.5f * pw, py2 = py + 0.5f * ph;
        const float ltx = fmaxf(px1, tx1), lty = fmaxf(py1, ty1);
        const float rbx = fminf(px2, tx2), rby = fminf(py2, ty2);
        const float iw = fmaxf(rbx - ltx, 0.0f);
        const float ih = fmaxf(rby - lty, 0.0f);
        const float inter = iw * ih;
        const float pa = (px2 - px1) * (py2 - py1);
        const float iou = inter / (pa + ta - inter);   // exact div: feeds loss
        if (b == 0) iou0 = iou; else iou1 = iou;
      }
      const int idx = (iou1 > iou0) ? 1 : 0;          // jnp.argmax: ties -> 0
      const float max_iou = fmaxf(iou0, iou1);
      const int ro = idx * 5;                          // responsible box offset
      const int no = 5 - ro;                           // not-responsible offset

      const float dx = p[ro + 0] - t[ro + 0];
      const float dy = p[ro + 1] - t[ro + 1];
      const float dw = sqrtf(p[ro + 2]) - sqrtf(t[ro + 2]);
      const float dh = sqrtf(p[ro + 3]) - sqrtf(t[ro + 3]);
      const float loc = dx * dx + dy * dy + dw * dw + dh * dh;

      const float dc = p[ro + 4] - max_iou;
      const float contain = dc * dc;
      const float ncf = p[no + 4];

      float cls = 0.0f;
      #pragma unroll
      for (int k = 10; k < 30; ++k) {
        const float d = p[k] - t[k];
        cls += d * d;
      }
      val += 5.0f * loc + 2.0f * contain + ncf * ncf + cls;
    }
  }

  // ---- deterministic block reduction (wave32 shuffle tree + LDS) ----
  #pragma unroll
  for (int off = 16; off > 0; off >>= 1)
    val += __shfl_down(val, off, 32);
  const int lane = tid & 31, wid = tid >> 5;
  if (lane == 0) red[wid] = val;
  __syncthreads();
  if (tid == 0) {
    float s = 0.0f;
    #pragma unroll
    for (int w = 0; w < BLOCK / 32; ++w) s += red[w];
    partials[blk] = s;                       // fixed slot per block: deterministic
  }
}

__global__ __launch_bounds__(256)
void yolo_loss_finalize(const float* __restrict__ partials, int n,
                        float* __restrict__ out, double nbatch) {
  __shared__ double red[8];
  double acc = 0.0;
  for (int i = threadIdx.x; i < n; i += 256)     // fixed order per thread
    acc += (double)partials[i];
  #pragma unroll
  for (int off = 16; off > 0; off >>= 1)
    acc += __shfl_down(acc, off, 32);
  const int lane = threadIdx.x & 31, wid = threadIdx.x >> 5;
  if (lane == 0) red[wid] = acc;
  __syncthreads();
  if (threadIdx.x == 0) {
    double s = 0.0;
    #pragma unroll
    for (int w = 0; w < 8; ++w) s += red[w];
    out[0] = (float)(s / nbatch);
  }
}

extern "C" void kernel_launch(void* const* d_in, const int* in_sizes, int n_in,
                              void* d_out, int out_size, void* d_ws, size_t ws_size,
                              hipStream_t stream) {
  const float* pred = (const float*)d_in[0];
  const float* targ = (const float*)d_in[1];
  const int total  = in_sizes[0];                           // N*14*14*30
  const int ncells = total / CELL_FLOATS;                   // N*196
  const int nbatch = ncells / (SGRID * SGRID);              // N
  const int blocks = (ncells + BLOCK - 1) / BLOCK;          // 3136 for N=4096

  float* partials = (float*)d_ws;                           // blocks * 4 bytes
  yolo_loss_stage1<<<blocks, BLOCK, 0, stream>>>(pred, targ, partials, ncells);
  yolo_loss_finalize<<<1, 256, 0, stream>>>(partials, blocks, (float*)d_out,
                                            (double)nbatch);
}